// YolinoLoss_19310172962887
// MI455X (gfx1250) — compile-verified
//
#include <hip/hip_runtime.h>
#include <math.h>

typedef __attribute__((ext_vector_type(2))) float v2f;
typedef __attribute__((ext_vector_type(8))) float v8f;

#define GH 96
#define GW 96
#define BATCH 8
#define TCH 64                      // target channels (16 lines x 4)
#define PCH 96                      // pred channels (16 preds x 6)
#define PLANE (GH * GW)             // 9216
#define CELLS (BATCH * GH * GW)     // 73728
#define WAVES_PER_BLOCK 8
#define NBLOCKS (CELLS / WAVES_PER_BLOCK)  // 9216, exact

// ds_swizzle_b32 group-of-32 XOR exchange: lane' = ((lane & 0x1f) | 0) ^ MASK.
// One DS instruction, no shuffle-address VALU arithmetic (vs ds_bpermute).
template <int MASK>
__device__ __forceinline__ float swz_f(float v) {
    return __int_as_float(
        __builtin_amdgcn_ds_swizzle(__float_as_int(v), (MASK << 10) | 0x1f));
}
template <int MASK>
__device__ __forceinline__ int swz_i(int v) {
    return __builtin_amdgcn_ds_swizzle(v, (MASK << 10) | 0x1f);
}

// One wave32 per grid cell. Distance matrix via two V_WMMA_F32_16X16X4_F32:
//   d^2[m,n] = x_m*(-2 px_n) + y_m*(-2 py_n) + |t_m|^2 * 1 + 1 * |p_n|^2
// A (16x4 f32): lanes 0-15 hold (K0,K1)=(x,y); lanes 16-31 hold (K2,K3)=(|t|^2, 1)
// B (4x16 f32): v0 = {K0 lanes0-15, K2 lanes16-31}, v1 = {K1, K3}
// C/D (16x16 f32): VGPR r: lanes 0-15 -> M=r, lanes 16-31 -> M=r+8; N = lane&15
__global__ __launch_bounds__(256) void yolino_cell_kernel(
    const float* __restrict__ target, const float* __restrict__ pred,
    float* __restrict__ partial)
{
    __shared__ float wsum[WAVES_PER_BLOCK];

    const int lane = threadIdx.x & 31;
    const int wv   = threadIdx.x >> 5;
    const int cell = blockIdx.x * WAVES_PER_BLOCK + wv;   // always < CELLS (exact fit)

    const int w  = cell % GW;
    const int hb = cell / GW;
    const int h  = hb % GH;
    const int b  = hb / GH;

    const long tbase = (long)b * TCH * PLANE + (long)h * GW + w;
    const long pbase = (long)b * PCH * PLANE + (long)h * GW + w;

    const int m    = lane & 15;     // both the target line and predictor this lane loads
    const int half = lane >> 4;     // 0: lanes 0-15, 1: lanes 16-31

    // ---- load target line m (4 channels, stride PLANE) ----
    const float tx1 = target[tbase + (long)(4 * m + 0) * PLANE];
    const float ty1 = target[tbase + (long)(4 * m + 1) * PLANE];
    const float tx2 = target[tbase + (long)(4 * m + 2) * PLANE];
    const float ty2 = target[tbase + (long)(4 * m + 3) * PLANE];
    const bool tline = (tx1 + ty1 + tx2 + ty2) > 0.0f;
    const unsigned tmask = (unsigned)__ballot(tline);     // bit i == tline[i&15]

    // ---- load predictor m (6 channels) ----
    const float px1 = pred[pbase + (long)(6 * m + 0) * PLANE];
    const float py1 = pred[pbase + (long)(6 * m + 1) * PLANE];
    const float px2 = pred[pbase + (long)(6 * m + 2) * PLANE];
    const float py2 = pred[pbase + (long)(6 * m + 3) * PLANE];
    const float cnf = pred[pbase + (long)(6 * m + 4) * PLANE];
    const float cls = pred[pbase + (long)(6 * m + 5) * PLANE];

    // ---- build WMMA operands (branch-free selects keep EXEC uniform) ----
    const bool hi = (half != 0);
    v2f A1, A2, B1, B2;
    A1.x = hi ? (tx1 * tx1 + ty1 * ty1) : tx1;
    A1.y = hi ? 1.0f                    : ty1;
    A2.x = hi ? (tx2 * tx2 + ty2 * ty2) : tx2;
    A2.y = hi ? 1.0f                    : ty2;
    B1.x = hi ? 1.0f                      : (-2.0f * px1);
    B1.y = hi ? (px1 * px1 + py1 * py1)   : (-2.0f * py1);
    B2.x = hi ? 1.0f                      : (-2.0f * px2);
    B2.y = hi ? (px2 * px2 + py2 * py2)   : (-2.0f * py2);

    v8f cz = {};
    v8f d1 = __builtin_amdgcn_wmma_f32_16x16x4_f32(false, A1, false, B1, (short)0, cz, false, false);
    v8f d2 = __builtin_amdgcn_wmma_f32_16x16x4_f32(false, A2, false, B2, (short)0, cz, false, false);

    // ---- per-pair distance + column (n = lane&15) min over rows ----
    // raw v_sqrt_f32 with |x| source modifier: tiny negative rounding error -> ~0
    float dist[8];
    float mn = 1e30f;
#pragma unroll
    for (int r = 0; r < 8; ++r) {
        dist[r] = __builtin_amdgcn_sqrtf(__builtin_fabsf(d1[r])) +
                  __builtin_amdgcn_sqrtf(__builtin_fabsf(d2[r]));
        mn = fminf(mn, dist[r]);
    }
    mn = fminf(mn, swz_f<16>(mn));            // combine M 0-7 with M 8-15
    const bool have = mn < 2.0f;              // dist >= 0, so thr==-1 never matches

    // ---- masked accumulation ----
    float part = 0.0f;
    int   cnt  = 0;
#pragma unroll
    for (int r = 0; r < 8; ++r) {
        const int  M   = r + (half << 3);
        const bool tl  = (tmask >> M) & 1u;
        const bool ijk = have && tl && (dist[r] == mn);
        if (ijk) { part += dist[r]; cnt += 1; }
    }
    // class term: (1 - (sigmoid(cls) > 0.5))^2 == (cls > 0 ? 0 : 1), per match
    if (!(cls > 0.0f)) part += (float)cnt;

    // conf term, once per predictor column (restrict to lanes 0-15)
    int anyc = (cnt > 0) ? 1 : 0;
    anyc |= swz_i<16>(anyc);
    if (half == 0) {
        const float s = __builtin_amdgcn_rcpf(1.0f + __expf(-cnf));
        part += anyc ? (s - 1.0f) * (s - 1.0f) : s * s;
    }

    // ---- deterministic wave reduction (ds_swizzle XOR butterfly) ----
    part += swz_f<16>(part);
    part += swz_f<8>(part);
    part += swz_f<4>(part);
    part += swz_f<2>(part);
    part += swz_f<1>(part);

    if (lane == 0) wsum[wv] = part;
    __syncthreads();
    if (threadIdx.x == 0) {
        float acc = 0.0f;
#pragma unroll
        for (int i = 0; i < WAVES_PER_BLOCK; ++i) acc += wsum[i];
        partial[blockIdx.x] = acc;
    }
}

// Single-block deterministic final reduction: fixed strided order + fixed tree.
__global__ __launch_bounds__(256) void yolino_reduce_kernel(
    const float* __restrict__ partial, float* __restrict__ out)
{
    __shared__ float sm[256];
    float acc = 0.0f;
    for (int i = threadIdx.x; i < NBLOCKS; i += 256) acc += partial[i];
    sm[threadIdx.x] = acc;
    __syncthreads();
    for (int s = 128; s > 0; s >>= 1) {
        if ((int)threadIdx.x < s) sm[threadIdx.x] += sm[threadIdx.x + s];
        __syncthreads();
    }
    if (threadIdx.x == 0) out[0] = sm[0] * (1.0f / (float)BATCH);
}

extern "C" void kernel_launch(void* const* d_in, const int* in_sizes, int n_in,
                              void* d_out, int out_size, void* d_ws, size_t ws_size,
                              hipStream_t stream) {
    const float* target = (const float*)d_in[0];  // (8, 64, 96, 96) fp32
    const float* pred   = (const float*)d_in[1];  // (8, 96, 96, 96) fp32
    float* partial = (float*)d_ws;                // NBLOCKS floats (36 KB)
    yolino_cell_kernel<<<NBLOCKS, 256, 0, stream>>>(target, pred, partial);
    yolino_reduce_kernel<<<1, 256, 0, stream>>>(partial, (float*)d_out);
}